// KLMinimizerThreshold_39101382263521
// MI455X (gfx1250) — compile-verified
//
#include <hip/hip_runtime.h>
#include <hip/hip_bf16.h>

// ---------------------------------------------------------------------------
// KL-divergence int8 calibration threshold search (MI455X / gfx1250).
// Phase 1/2 are HBM/L2 streaming (roofline ~11.5us @ 23.3TB/s; 134MB fits the
// 192MB L2 so pass 2 mostly hits L2). Phase 3 is 438 tiny per-candidate KL
// evaluations. CDNA5 paths: async global->LDS staging (ASYNCcnt) in the
// histogram pass, v_wmma_f32_16x16x4_f32 wave reductions in the KL pass.
// ---------------------------------------------------------------------------

#define NBINS   1001
#define KQ      127
#define HALFB   500
#define EPS_S   1e-4f
#define NCAND   438      // i = 63 .. 500

typedef __attribute__((ext_vector_type(2))) float v2f;
typedef __attribute__((ext_vector_type(8))) float v8f;

// ---- wave32 sum reduction through the WMMA pipe ---------------------------
// A[m][0] = v[lane m], A[m][2] = v[lane m+16] (ISA 16x4 A layout), B = ones
// => D[m][n] = v[m] + v[m+16] for all n.  Sum the 8 D VGPRs per lane:
// lanes 0-15 get rows 0..7, lanes 16-31 rows 8..15; one shfl_xor(16) finishes.
__device__ __forceinline__ float wave_sum_wmma(float v) {
  v2f a = {v, 0.0f};
  v2f b = {1.0f, 1.0f};
  v8f c = {};
  v8f d = __builtin_amdgcn_wmma_f32_16x16x4_f32(
      /*neg_a=*/false, a, /*neg_b=*/false, b,
      /*c_mod=*/(short)0, c, /*reuse_a=*/false, /*reuse_b=*/false);
  float s = 0.0f;
#pragma unroll
  for (int i = 0; i < 8; ++i) s += d[i];
  s += __shfl_xor(s, 16, 32);
  return s;
}

__device__ __forceinline__ float wave_max32(float v) {
#pragma unroll
  for (int off = 16; off > 0; off >>= 1)
    v = fmaxf(v, __shfl_xor(v, off, 32));
  return v;
}

template <int NT>
__device__ __forceinline__ float block_sum(float v, float* red) {
  float w = wave_sum_wmma(v);            // uniform call-site: EXEC all ones
  __syncthreads();
  if ((threadIdx.x & 31) == 0) red[threadIdx.x >> 5] = w;
  __syncthreads();
  float t = 0.0f;
#pragma unroll
  for (int i = 0; i < NT / 32; ++i) t += red[i];
  return t;
}

template <int NT>
__device__ __forceinline__ float block_max(float v, float* red) {
  float w = wave_max32(v);
  __syncthreads();
  if ((threadIdx.x & 31) == 0) red[threadIdx.x >> 5] = w;
  __syncthreads();
  float t = -3.0e38f;
#pragma unroll
  for (int i = 0; i < NT / 32; ++i) t = fmaxf(t, red[i]);
  return t;
}

// ---- async global -> LDS (CDNA5 VGLOBAL async path, tracked by ASYNCcnt) --
__device__ __forceinline__ void async_copy_b128(unsigned lds_off, const void* g) {
  asm volatile("global_load_async_to_lds_b128 %0, %1, off"
               :: "v"(lds_off), "v"(g) : "memory");
}
__device__ __forceinline__ void wait_async_le1() {
  asm volatile("s_wait_asynccnt 0x1" ::: "memory");
}
__device__ __forceinline__ void wait_async_le0() {
  asm volatile("s_wait_asynccnt 0x0" ::: "memory");
}

// ---------------------------------------------------------------------------
__global__ void init_ws_kernel(unsigned* amax_bits, unsigned* ghist) {
  int t = blockIdx.x * blockDim.x + threadIdx.x;
  if (t == 0) *amax_bits = 0u;
  if (t < NBINS) ghist[t] = 0u;
}

// Pass 1: absmax.  RT caching on purpose: 134MB < 192MB L2, so pass 2 can hit L2.
__global__ void absmax_kernel(const float* __restrict__ x, unsigned* amax_bits,
                              int n4, int n) {
  int gid = blockIdx.x * blockDim.x + threadIdx.x;
  int stride = gridDim.x * blockDim.x;
  const float4* x4 = (const float4*)x;
  float m = 0.0f;
  for (int i = gid; i < n4; i += stride) {
    float4 v = x4[i];
    m = fmaxf(m, fmaxf(fmaxf(fabsf(v.x), fabsf(v.y)),
                       fmaxf(fabsf(v.z), fabsf(v.w))));
  }
  for (int i = 4 * n4 + gid; i < n; i += stride)   // tail (none for 2^25)
    m = fmaxf(m, fabsf(x[i]));
  m = wave_max32(m);
  if ((threadIdx.x & 31) == 0)
    atomicMax(amax_bits, __float_as_uint(m));      // |x| bits are order-preserving
}

// Pass 2: histogram with double-buffered async LDS staging + LDS atomics.
#define HBLK 256
__global__ void hist_kernel(const float* __restrict__ x,
                            const unsigned* __restrict__ amax_bits,
                            unsigned* __restrict__ ghist,
                            int nChunks, int n) {
  __shared__ __align__(16) float tile[2][HBLK * 4];
  __shared__ unsigned lhist[NBINS];
  const int tid = threadIdx.x;

  for (int j = tid; j < NBINS; j += HBLK) lhist[j] = 0u;
  __syncthreads();

  const float am = __uint_as_float(*amax_bits);
  const float inv_w = (float)NBINS / (2.0f * am);

  const unsigned ldsA = (unsigned)(size_t)&tile[0][tid * 4];
  const unsigned ldsB = (unsigned)(size_t)&tile[1][tid * 4];

  int c = blockIdx.x;
  if (c < nChunks)
    async_copy_b128(ldsA, x + (size_t)c * (HBLK * 4) + tid * 4);

  int buf = 0;
  for (; c < nChunks; c += gridDim.x) {
    int cn = c + gridDim.x;
    if (cn < nChunks) {
      async_copy_b128(buf ? ldsA : ldsB,
                      x + (size_t)cn * (HBLK * 4) + tid * 4);
      wait_async_le1();                 // in-order completion frees older buffer
    } else {
      wait_async_le0();
    }
    float4 v = *(const float4*)&tile[buf][tid * 4];
    int b0 = (int)((v.x + am) * inv_w); if (b0 > NBINS - 1) b0 = NBINS - 1; if (b0 < 0) b0 = 0;
    int b1 = (int)((v.y + am) * inv_w); if (b1 > NBINS - 1) b1 = NBINS - 1; if (b1 < 0) b1 = 0;
    int b2 = (int)((v.z + am) * inv_w); if (b2 > NBINS - 1) b2 = NBINS - 1; if (b2 < 0) b2 = 0;
    int b3 = (int)((v.w + am) * inv_w); if (b3 > NBINS - 1) b3 = NBINS - 1; if (b3 < 0) b3 = 0;
    atomicAdd(&lhist[b0], 1u);
    atomicAdd(&lhist[b1], 1u);
    atomicAdd(&lhist[b2], 1u);
    atomicAdd(&lhist[b3], 1u);
    buf ^= 1;
  }

  if (blockIdx.x == 0) {                // tail elements (none for 2^25)
    for (int i = nChunks * (HBLK * 4) + tid; i < n; i += HBLK) {
      float v = x[i];
      int b = (int)((v + am) * inv_w); if (b > NBINS - 1) b = NBINS - 1; if (b < 0) b = 0;
      atomicAdd(&lhist[b], 1u);
    }
  }
  __syncthreads();
  for (int j = tid; j < NBINS; j += HBLK) {
    unsigned cv = lhist[j];
    if (cv) atomicAdd(&ghist[j], cv);
  }
}

// Pass 3: one block per candidate i = 63+blockIdx.x.
#define KBLK 256
__global__ void kl_kernel(const unsigned* __restrict__ ghist,
                          const unsigned* __restrict__ amax_bits,
                          float* __restrict__ divg, float* __restrict__ thr) {
  __shared__ float sh[NBINS];
  __shared__ float qb[KQ];
  __shared__ float nrm[KQ];
  __shared__ float red[KBLK / 32];
  const int tid = threadIdx.x;
  const int i = 63 + blockIdx.x;
  const int start = HALFB - i, stop = HALFB + i + 1, L = 2 * i + 1;

  for (int j = tid; j < NBINS; j += KBLK) sh[j] = (float)ghist[j];
  if (tid < KQ) { qb[tid] = 0.0f; nrm[tid] = 0.0f; }
  __syncthreads();

  // tail masses folded into p[0] / p[L-1]
  float ls = 0.0f, rs = 0.0f;
  for (int j = tid; j < start; j += KBLK) ls += sh[j];
  for (int j = stop + tid; j < NBINS; j += KBLK) rs += sh[j];
  const float left  = block_sum<KBLK>(ls, red);
  const float right = block_sum<KBLK>(rs, red);

  const int m = L / KQ;                 // merged-bin width
  for (int k = tid; k < L; k += KBLK) { // segment sums + nonzero counts
    float v = sh[start + k];
    if (v != 0.0f) {
      int seg = k / m; if (seg > KQ - 1) seg = KQ - 1;
      atomicAdd(&qb[seg], v);
      if (k != L - 1) atomicAdd(&nrm[seg], 1.0f);
    }
  }
  __syncthreads();

  // smoothing stats for p and q
  float zp = 0.0f, zq = 0.0f;
  for (int k = tid; k < L; k += KBLK) {
    float v = sh[start + k];
    float p = v + (k == 0 ? left : 0.0f) + (k == L - 1 ? right : 0.0f);
    int seg = k / m; if (seg > KQ - 1) seg = KQ - 1;
    float nk = nrm[seg];
    float q = (v != 0.0f && k != L - 1 && nk > 0.0f) ? qb[seg] / fmaxf(nk, 1.0f) : 0.0f;
    if (p == 0.0f) zp += 1.0f;
    if (q == 0.0f) zq += 1.0f;
  }
  const float n_zeros_p = block_sum<KBLK>(zp, red);
  const float n_zeros_q = block_sum<KBLK>(zq, red);
  const float Lf = (float)L;
  const float n_nz_p = Lf - n_zeros_p;
  const float n_nz_q = Lf - n_zeros_q;
  const float off_p = (EPS_S * n_zeros_p / fmaxf(n_nz_p, 1.0f)) * n_nz_p;
  const float off_q = (EPS_S * n_zeros_q / fmaxf(n_nz_q, 1.0f)) * n_nz_q;

  // logsumexp: max pass
  float mp = -3.0e38f, mq = -3.0e38f;
  for (int k = tid; k < L; k += KBLK) {
    float v = sh[start + k];
    float p = v + (k == 0 ? left : 0.0f) + (k == L - 1 ? right : 0.0f);
    int seg = k / m; if (seg > KQ - 1) seg = KQ - 1;
    float nk = nrm[seg];
    float q = (v != 0.0f && k != L - 1 && nk > 0.0f) ? qb[seg] / fmaxf(nk, 1.0f) : 0.0f;
    float plog = p + (p == 0.0f ? EPS_S : 0.0f) - off_p;
    float qlog = q + (q == 0.0f ? EPS_S : 0.0f) - off_q;
    mp = fmaxf(mp, plog);
    mq = fmaxf(mq, qlog);
  }
  mp = block_max<KBLK>(mp, red);
  mq = block_max<KBLK>(mq, red);

  // logsumexp: sum pass
  float sp = 0.0f, sq = 0.0f;
  for (int k = tid; k < L; k += KBLK) {
    float v = sh[start + k];
    float p = v + (k == 0 ? left : 0.0f) + (k == L - 1 ? right : 0.0f);
    int seg = k / m; if (seg > KQ - 1) seg = KQ - 1;
    float nk = nrm[seg];
    float q = (v != 0.0f && k != L - 1 && nk > 0.0f) ? qb[seg] / fmaxf(nk, 1.0f) : 0.0f;
    float plog = p + (p == 0.0f ? EPS_S : 0.0f) - off_p;
    float qlog = q + (q == 0.0f ? EPS_S : 0.0f) - off_q;
    sp += __expf(plog - mp);
    sq += __expf(qlog - mq);
  }
  sp = block_sum<KBLK>(sp, red);
  sq = block_sum<KBLK>(sq, red);
  const float lZp = mp + __logf(sp);
  const float lZq = mq + __logf(sq);

  // KL sum
  float klv = 0.0f;
  for (int k = tid; k < L; k += KBLK) {
    float v = sh[start + k];
    float p = v + (k == 0 ? left : 0.0f) + (k == L - 1 ? right : 0.0f);
    int seg = k / m; if (seg > KQ - 1) seg = KQ - 1;
    float nk = nrm[seg];
    float q = (v != 0.0f && k != L - 1 && nk > 0.0f) ? qb[seg] / fmaxf(nk, 1.0f) : 0.0f;
    float plog = p + (p == 0.0f ? EPS_S : 0.0f) - off_p;
    float qlog = q + (q == 0.0f ? EPS_S : 0.0f) - off_q;
    float lp = plog - lZp;
    klv += __expf(lp) * (lp - (qlog - lZq));
  }
  float kl = block_sum<KBLK>(klv, red);

  if (tid == 0) {
    if (!(n_nz_q > 0.0f)) kl = INFINITY;
    divg[blockIdx.x] = kl;
    float am = __uint_as_float(*amax_bits);
    thr[blockIdx.x] = -am + (float)stop * (2.0f * am / (float)NBINS);  // edges[stop]
  }
}

__global__ void argmin_kernel(const float* __restrict__ divg,
                              const float* __restrict__ thr,
                              float* __restrict__ out) {
  __shared__ float sval[256];
  __shared__ int   sidx[256];
  const int tid = threadIdx.x;
  float best = INFINITY; int bi = 0x7fffffff;
  for (int j = tid; j < NCAND; j += 256) {
    float d = divg[j];
    if (d < best || (d == best && j < bi)) { best = d; bi = j; }
  }
  sval[tid] = best; sidx[tid] = bi;
  __syncthreads();
  for (int s = 128; s > 0; s >>= 1) {
    if (tid < s) {
      float o = sval[tid + s]; int oi = sidx[tid + s];
      if (o < sval[tid] || (o == sval[tid] && oi < sidx[tid])) {
        sval[tid] = o; sidx[tid] = oi;
      }
    }
    __syncthreads();
  }
  if (tid == 0) out[0] = thr[sidx[0]];
}

// ---------------------------------------------------------------------------
extern "C" void kernel_launch(void* const* d_in, const int* in_sizes, int n_in,
                              void* d_out, int out_size, void* d_ws, size_t ws_size,
                              hipStream_t stream) {
  const float* x = (const float*)d_in[0];
  const int n = in_sizes[0];

  // workspace layout (needs < 8KB)
  unsigned* amax  = (unsigned*)d_ws;                        // 4B
  unsigned* ghist = (unsigned*)((char*)d_ws + 16);          // 1001*4B
  float*    divg  = (float*)((char*)d_ws + 4096);           // 438*4B
  float*    thr   = (float*)((char*)d_ws + 6144);           // 438*4B
  float*    out   = (float*)d_out;

  hipLaunchKernelGGL(init_ws_kernel, dim3(1), dim3(1024), 0, stream, amax, ghist);

  const int n4 = n / 4;
  hipLaunchKernelGGL(absmax_kernel, dim3(2048), dim3(256), 0, stream,
                     x, amax, n4, n);

  const int nChunks = n / (HBLK * 4);
  hipLaunchKernelGGL(hist_kernel, dim3(2048), dim3(HBLK), 0, stream,
                     x, amax, ghist, nChunks, n);

  hipLaunchKernelGGL(kl_kernel, dim3(NCAND), dim3(KBLK), 0, stream,
                     ghist, amax, divg, thr);

  hipLaunchKernelGGL(argmin_kernel, dim3(1), dim3(256), 0, stream,
                     divg, thr, out);
}